// TaxaNetLoss_34952443855389
// MI455X (gfx1250) — compile-verified
//
#include <hip/hip_runtime.h>
#include <stdint.h>

// ---- problem constants (from reference) ------------------------------------
#define N_SAMPLES 1024
#define NCLS      13430          // PIQUETS[-1]
#define L1_BEG    30             // PIQUETS[1]
#define L2_BEG    430            // PIQUETS[2]
#define L3_BEG    3430           // PIQUETS[3]
#define ROWPAD    13432          // row padded to 16B multiple

// ---- CDNA5 async-to-LDS helpers (builtin if present, inline asm fallback) --
typedef int v2i __attribute__((ext_vector_type(2)));
typedef __attribute__((address_space(1))) v2i glb_v2i;
typedef __attribute__((address_space(3))) v2i lds_v2i;

__device__ __forceinline__ void async_ld_b64(const float* gsrc, float* ldst) {
#if defined(__AMDGCN__) && __has_builtin(__builtin_amdgcn_global_load_async_to_lds_b64)
  __builtin_amdgcn_global_load_async_to_lds_b64(
      (glb_v2i*)(uintptr_t)gsrc,
      (lds_v2i*)(unsigned)(uintptr_t)ldst,
      0, 0);
#else
  asm volatile("global_load_async_to_lds_b64 %0, %1, off"
               :: "v"((unsigned)(uintptr_t)ldst), "v"(gsrc)
               : "memory");
#endif
}

__device__ __forceinline__ void wait_async0() {
#if defined(__AMDGCN__) && __has_builtin(__builtin_amdgcn_s_wait_asynccnt)
  __builtin_amdgcn_s_wait_asynccnt(0);
#else
  asm volatile("s_wait_asynccnt 0x0" ::: "memory");
#endif
}

// ---- order-preserving (value, index) packing for u64 max reduction ---------
// key = (monotonic_u32(value) << 32) | ~index  -> u64 max picks larger value,
// ties pick the SMALLER index (first occurrence), matching jnp.argmax.
__device__ __forceinline__ unsigned long long packkey(float v, unsigned idx) {
  unsigned u = __float_as_uint(v);
  u = (u & 0x80000000u) ? ~u : (u | 0x80000000u);
  return ((unsigned long long)u << 32) | (unsigned)(~idx);
}
__device__ __forceinline__ float keyval(unsigned long long k) {
  unsigned u = (unsigned)(k >> 32);
  u = (u & 0x80000000u) ? (u & 0x7FFFFFFFu) : ~u;
  return __uint_as_float(u);
}
__device__ __forceinline__ int keyidx(unsigned long long k) {
  return (int)(~(unsigned)k);
}

// ---- Kernel A: one block per sample ----------------------------------------
__global__ __launch_bounds__(256)
void taxa_rows(const float* __restrict__ y_pred,
               const int*   __restrict__ y_true,
               int*         __restrict__ argm,   // [4][N]
               float*       __restrict__ ce) {   // [3][N]
  __shared__ __attribute__((aligned(16))) float row[ROWPAD];
  __shared__ unsigned long long slot[4];
  __shared__ float fred[3 * 256];

  const int tid = threadIdx.x;
  const int i   = blockIdx.x;
  const float* g = y_pred + (size_t)i * NCLS;

  // Stage the whole row into LDS via the async data mover (b64: row stride
  // 53720B is 8B- but not 16B-aligned). 6715 chunks, no tail.
  for (int c = tid; c < NCLS / 2; c += 256)
    async_ld_b64(g + 2 * c, row + 2 * c);
  wait_async0();

  // Seed each level's argmax slot with the out-of-level zero candidate:
  // masked logits are literal 0.0 outside the level; the first such index is
  // 30 for level 0 (level starts at 0) and 0 for levels 1..3.
  if (tid < 4) {
    const unsigned z = (tid == 0) ? (unsigned)L1_BEG : 0u;
    slot[tid] = packkey(0.0f, z);
  }
  __syncthreads();

  // Pass 1: per-thread in-level argmax (strict > keeps first occurrence
  // within a thread's increasing stride sequence).
  const float NINF = -__builtin_inff();
  float m0 = NINF, m1 = NINF, m2 = NINF, m3 = NINF;
  unsigned i0 = 0x7fffffffu, i1 = i0, i2 = i0, i3 = i0;
  for (int j = tid; j < NCLS; j += 256) {
    float v = row[j];
    if (j < L1_BEG)      { if (v > m0) { m0 = v; i0 = (unsigned)j; } }
    else if (j < L2_BEG) { if (v > m1) { m1 = v; i1 = (unsigned)j; } }
    else if (j < L3_BEG) { if (v > m2) { m2 = v; i2 = (unsigned)j; } }
    else                 { if (v > m3) { m3 = v; i3 = (unsigned)j; } }
  }
  // Block-wide argmax via LDS u64 max atomics (ds_max_u64): exact integer
  // max => order-independent => deterministic.
  atomicMax(&slot[0], packkey(m0, i0));
  atomicMax(&slot[1], packkey(m1, i1));
  atomicMax(&slot[2], packkey(m2, i2));
  atomicMax(&slot[3], packkey(m3, i3));
  __syncthreads();

  // M_k = max over the MASKED row = max(in-level max, 0) -> value part of key.
  const float M1 = keyval(slot[1]);
  const float M2 = keyval(slot[2]);
  const float M3 = keyval(slot[3]);

  // Pass 2: in-level sum of exp(v - M_k) from LDS (fixed per-thread order).
  float s1 = 0.f, s2 = 0.f, s3 = 0.f;
  for (int j = L1_BEG + tid; j < L2_BEG; j += 256) s1 += expf(row[j] - M1);
  for (int j = L2_BEG + tid; j < L3_BEG; j += 256) s2 += expf(row[j] - M2);
  for (int j = L3_BEG + tid; j < NCLS;   j += 256) s3 += expf(row[j] - M3);
  fred[tid] = s1; fred[256 + tid] = s2; fred[512 + tid] = s3;
  __syncthreads();
  for (int s = 128; s > 0; s >>= 1) {          // deterministic tree reduction
    if (tid < s) {
      fred[tid]       += fred[tid + s];
      fred[256 + tid] += fred[256 + tid + s];
      fred[512 + tid] += fred[512 + tid + s];
    }
    __syncthreads();
  }

  if (tid == 0) {
    // Out-of-level masked zeros contribute (C - L_k) * exp(-M_k) each.
    const float S1 = fred[0]   + (float)(NCLS -   400) * expf(-M1);
    const float S2 = fred[256] + (float)(NCLS -  3000) * expf(-M2);
    const float S3 = fred[512] + (float)(NCLS - 10000) * expf(-M3);

    argm[0 * N_SAMPLES + i] = keyidx(slot[0]);
    argm[1 * N_SAMPLES + i] = keyidx(slot[1]);
    argm[2 * N_SAMPLES + i] = keyidx(slot[2]);
    argm[3 * N_SAMPLES + i] = keyidx(slot[3]);

    const int t1 = y_true[i * 4 + 1];
    const int t2 = y_true[i * 4 + 2];
    const int t3 = y_true[i * 4 + 3];
    // -log_softmax[t] = lse - logit[t]
    ce[0 * N_SAMPLES + i] = (M1 + logf(S1)) - row[t1];
    ce[1 * N_SAMPLES + i] = (M2 + logf(S2)) - row[t2];
    ce[2 * N_SAMPLES + i] = (M3 + logf(S3)) - row[t3];
  }
}

// ---- Kernel B: combine penalties + CE means into the scalar loss -----------
__global__ __launch_bounds__(256)
void taxa_final(const int*   __restrict__ argm,  // [4][N]
                const float* __restrict__ ce,    // [3][N]
                const float* __restrict__ H,     // [C][C]
                float*       __restrict__ out) {
  __shared__ int   ired[256];
  __shared__ float fred[256];
  const int tid = threadIdx.x;
  const float E = 2.71828182845904523536f;
  const float W[4] = {0.25f, 0.25f, 0.15f, 0.10f};

  float loss = 0.f;
  for (int k = 1; k < 4; ++k) {
    // hierarchy-violation count over samples 1..N-1 (reference skips i=0)
    int cnt = 0;
    for (int i = 1 + tid; i < N_SAMPLES; i += 256) {
      const int a = argm[(k - 1) * N_SAMPLES + i];
      const int b = argm[k * N_SAMPLES + i];
      cnt += (H[(size_t)a * NCLS + b] == 0.0f) ? 1 : 0;
    }
    ired[tid] = cnt;
    // CE partial sum over all N samples (fixed stride order -> deterministic)
    float cs = 0.f;
    for (int i = tid; i < N_SAMPLES; i += 256)
      cs += ce[(k - 1) * N_SAMPLES + i];
    fred[tid] = cs;
    __syncthreads();
    for (int s = 128; s > 0; s >>= 1) {
      if (tid < s) { ired[tid] += ired[tid + s]; fred[tid] += fred[tid + s]; }
      __syncthreads();
    }
    loss += W[k] * ((float)ired[0] * E + fred[0] / (float)N_SAMPLES);
    __syncthreads();
  }
  if (tid == 0) out[0] = loss;
}

// ---- host entry ------------------------------------------------------------
extern "C" void kernel_launch(void* const* d_in, const int* in_sizes, int n_in,
                              void* d_out, int out_size, void* d_ws, size_t ws_size,
                              hipStream_t stream) {
  const float* y_pred = (const float*)d_in[0];   // [1024, 13430] f32
  const int*   y_true = (const int*)  d_in[1];   // [1024, 4] i32
  const float* H      = (const float*)d_in[2];   // [13430, 13430] f32

  int*   argm = (int*)d_ws;                                          // 16 KB
  float* ce   = (float*)((char*)d_ws + sizeof(int) * 4 * N_SAMPLES); // 12 KB

  taxa_rows<<<dim3(N_SAMPLES), dim3(256), 0, stream>>>(y_pred, y_true, argm, ce);
  taxa_final<<<dim3(1), dim3(256), 0, stream>>>(argm, ce, H, (float*)d_out);
}